// BCE_topK_loss_landmark_80814104641892
// MI455X (gfx1250) — compile-verified
//
#include <hip/hip_runtime.h>
#include <cstdint>
#include <cstddef>

// Problem constants (from the reference)
#define BATCH  2
#define CHN    8
#define DDIM   64
#define HDIM   192
#define WDIM   192
#define ROWLEN (DDIM*HDIM*WDIM)        // 2359296 voxels per (b,c) row
#define NROWS  (BATCH*CHN)             // 16 rows
#define NTOP   235930                  // round(ROWLEN * 0.10)
#define NBIN   4096
#define PATCH  15

#define THREADS        256
#define F4_PER_THREAD  16
#define F4_PER_BLOCK   (THREADS*F4_PER_THREAD)     // 4096 float4 per block
#define BLOCKS_PER_ROW ((ROWLEN/4)/F4_PER_BLOCK)   // 144 (exact, no tail)

// ---- gfx1250-specific feature probes (compile-safe) ----
#if defined(__HIP_DEVICE_COMPILE__) && __has_builtin(__builtin_amdgcn_global_load_async_to_lds_b128) && __has_builtin(__builtin_amdgcn_s_wait_asynccnt)
#define ATHENA_ASYNC_LDS 1
#else
#define ATHENA_ASYNC_LDS 0
#endif

#if defined(__HIP_DEVICE_COMPILE__) && __has_builtin(__builtin_amdgcn_wmma_f32_16x16x4_f32)
#define ATHENA_WMMA 1
#else
#define ATHENA_WMMA 0
#endif

typedef __attribute__((ext_vector_type(4))) int athena_v4i;

#if ATHENA_ASYNC_LDS
// Probe-derived prototype: (int4 AS1* gsrc, int4 AS3* ldst, imm offset, imm cpol)
__device__ __forceinline__ void athena_async16(const float* g, float* l) {
  __builtin_amdgcn_global_load_async_to_lds_b128(
      (__attribute__((address_space(1))) athena_v4i*)(g),
      (__attribute__((address_space(3))) athena_v4i*)(l), 0, 0);
}
#endif

// BCE-with-logits loss for 4 consecutive voxels (a float4 never crosses an H
// row since 192 % 4 == 0). Fixed op sequence (fmaf/add) so every pass computes
// bit-identical values.
__device__ __forceinline__ void athena_loss4(
    float4 v, int q, int d0, int h0, int w0,
    const float* __restrict__ tpatch, float lv[4], uint32_t lb[4])
{
  const int QHW = (HDIM*WDIM)/4;   // 9216
  const int QW  = WDIM/4;          // 48
  int d   = q / QHW;
  int rem = q - d*QHW;
  int h   = rem / QW;
  int wq  = (rem - h*QW) * 4;
  int du = d - d0, hu = h - h0;
  bool dh_in = ((unsigned)du < PATCH) & ((unsigned)hu < PATCH);
  float x4[4] = {v.x, v.y, v.z, v.w};
#pragma unroll
  for (int j = 0; j < 4; ++j) {
    float t = 0.f;
    int wu = wq + j - w0;
    if (dh_in & ((unsigned)wu < PATCH))
      t = tpatch[(du*PATCH + hu)*PATCH + wu];
    float x = x4[j];
    float l = __fmaf_rn(x, -t, fmaxf(x, 0.f)) + log1pf(__expf(-fabsf(x)));
    lv[j] = l;
    lb[j] = __float_as_uint(l);   // loss >= 0 -> bit order == value order
  }
}

// ---- Histogram pass (recompute variant). LEVEL 0/1/2 radix select.
// WB: also store the 32-bit loss patterns (L2-resident cache for later passes).
template<int LEVEL, bool WB>
__global__ void __launch_bounds__(THREADS)
athena_hist_kernel(const float* __restrict__ net,
                   const float* __restrict__ tstruct,
                   const int*   __restrict__ bboxes,
                   unsigned int* __restrict__ hist,
                   const unsigned int* __restrict__ state,
                   unsigned int* __restrict__ bits_out)
{
  __shared__ unsigned int lh[NBIN];
#if ATHENA_ASYNC_LDS
  __shared__ float tile[2][THREADS*4];
#endif
  const int r = blockIdx.y;
  const int d0 = bboxes[r*3+0];
  const int h0 = bboxes[r*3+1];
  const int w0 = bboxes[r*3+2];
  const float* tpatch = tstruct + (size_t)(r >> 3) * (PATCH*PATCH*PATCH);
  for (int i = threadIdx.x; i < NBIN; i += THREADS) lh[i] = 0u;
  unsigned int prefix = 0u;
  if (LEVEL != 0) prefix = state[r*4+0];
  __syncthreads();

  const float* rowbase = net + (size_t)r * ROWLEN;
  const int qbase = blockIdx.x * F4_PER_BLOCK + threadIdx.x;

#if ATHENA_ASYNC_LDS
  athena_async16(rowbase + (size_t)qbase*4,             &tile[0][threadIdx.x*4]);
  athena_async16(rowbase + (size_t)(qbase + THREADS)*4, &tile[1][threadIdx.x*4]);
#endif
  for (int it = 0; it < F4_PER_THREAD; ++it) {
    const int q = qbase + it*THREADS;
    float4 v;
#if ATHENA_ASYNC_LDS
    if (it + 1 < F4_PER_THREAD) __builtin_amdgcn_s_wait_asynccnt(1);
    else                        __builtin_amdgcn_s_wait_asynccnt(0);
    __syncthreads();
    v = ((const float4*)tile[it & 1])[threadIdx.x];
#else
    v = ((const float4*)rowbase)[q];
    if (it + 2 < F4_PER_THREAD)
      __builtin_prefetch(rowbase + (size_t)(q + 2*THREADS)*4, 0, 1);
#endif
    float lv[4]; uint32_t lb[4];
    athena_loss4(v, q, d0, h0, w0, tpatch, lv, lb);
    if (WB) {
      uint4 u; u.x = lb[0]; u.y = lb[1]; u.z = lb[2]; u.w = lb[3];
      ((uint4*)(bits_out + (size_t)r*ROWLEN))[q] = u;
    }
#pragma unroll
    for (int j = 0; j < 4; ++j) {
      unsigned int bits = lb[j];
      if (LEVEL == 0) {
        atomicAdd(&lh[bits >> 20], 1u);
      } else if (LEVEL == 1) {
        if ((bits >> 20) == (prefix >> 20)) atomicAdd(&lh[(bits >> 8) & 0xFFFu], 1u);
      } else {
        if ((bits >> 8) == (prefix >> 8)) atomicAdd(&lh[bits & 0xFFu], 1u);
      }
    }
#if ATHENA_ASYNC_LDS
    __syncthreads();
    if (it + 2 < F4_PER_THREAD)
      athena_async16(rowbase + (size_t)(q + 2*THREADS)*4, &tile[it & 1][threadIdx.x*4]);
#endif
  }
  __syncthreads();
  for (int i = threadIdx.x; i < NBIN; i += THREADS) {
    unsigned int c = lh[i];
    if (c) atomicAdd(hist + (size_t)r*NBIN + i, c);
  }
}

// ---- Histogram pass reading cached bit patterns (L2-resident: 151MB < 192MB L2)
template<int LEVEL>
__global__ void __launch_bounds__(THREADS)
athena_hist_bits_kernel(const unsigned int* __restrict__ bits_arr,
                        unsigned int* __restrict__ hist,
                        const unsigned int* __restrict__ state)
{
  __shared__ unsigned int lh[NBIN];
  const int r = blockIdx.y;
  for (int i = threadIdx.x; i < NBIN; i += THREADS) lh[i] = 0u;
  const unsigned int prefix = state[r*4+0];
  __syncthreads();
  const uint4* base = (const uint4*)(bits_arr + (size_t)r*ROWLEN);
  const int qbase = blockIdx.x * F4_PER_BLOCK + threadIdx.x;
  for (int it = 0; it < F4_PER_THREAD; ++it) {
    uint4 u = base[qbase + it*THREADS];
    unsigned int b4[4] = {u.x, u.y, u.z, u.w};
#pragma unroll
    for (int j = 0; j < 4; ++j) {
      unsigned int bits = b4[j];
      if (LEVEL == 1) {
        if ((bits >> 20) == (prefix >> 20)) atomicAdd(&lh[(bits >> 8) & 0xFFFu], 1u);
      } else {
        if ((bits >> 8) == (prefix >> 8)) atomicAdd(&lh[bits & 0xFFu], 1u);
      }
    }
  }
  __syncthreads();
  for (int i = threadIdx.x; i < NBIN; i += THREADS) {
    unsigned int c = lh[i];
    if (c) atomicAdd(hist + (size_t)r*NBIN + i, c);
  }
}

// ---- Per-row serial suffix scan of the histogram; picks radix digit, zeroes hist.
__global__ void athena_scan_kernel(unsigned int* __restrict__ hist,
                                   unsigned int* __restrict__ state, int level)
{
  const int r = blockIdx.x;
  unsigned int* h  = hist + (size_t)r*NBIN;
  unsigned int* st = state + r*4;
  if (threadIdx.x == 0) {
    unsigned int k = (level == 0) ? (unsigned int)NTOP : st[1];
    int nb = (level == 2) ? 256 : NBIN;
    unsigned int c = 0, sel = 0, cab = 0;
    for (int i = nb - 1; i >= 0; --i) {
      unsigned int hb = h[i];
      if (c + hb >= k) { sel = (unsigned int)i; cab = c; break; }
      c += hb;
    }
    unsigned int krem = k - cab;          // how many still needed inside bin sel
    if (level == 0)      { st[0] = sel << 20; st[1] = krem; }
    else if (level == 1) { st[0] |= sel << 8; st[1] = krem; }
    else                 { st[2] = st[0] | sel; st[1] = krem; }  // thr bits, k4
  }
  __syncthreads();
  for (int i = threadIdx.x; i < NBIN; i += blockDim.x) h[i] = 0u;
}

// ---- Final sum of values strictly above threshold (recompute variant)
__global__ void __launch_bounds__(THREADS)
athena_sum_kernel(const float* __restrict__ net,
                  const float* __restrict__ tstruct,
                  const int*   __restrict__ bboxes,
                  const unsigned int* __restrict__ state,
                  double* __restrict__ row_sum)
{
#if ATHENA_ASYNC_LDS
  __shared__ float tile[2][THREADS*4];
#endif
  __shared__ double sred[THREADS];
  const int r = blockIdx.y;
  const int d0 = bboxes[r*3+0], h0 = bboxes[r*3+1], w0 = bboxes[r*3+2];
  const float* tpatch = tstruct + (size_t)(r >> 3) * (PATCH*PATCH*PATCH);
  const unsigned int thr = state[r*4+2];
  const float* rowbase = net + (size_t)r * ROWLEN;
  const int qbase = blockIdx.x * F4_PER_BLOCK + threadIdx.x;
  double acc = 0.0;
#if ATHENA_ASYNC_LDS
  athena_async16(rowbase + (size_t)qbase*4,             &tile[0][threadIdx.x*4]);
  athena_async16(rowbase + (size_t)(qbase + THREADS)*4, &tile[1][threadIdx.x*4]);
#endif
  for (int it = 0; it < F4_PER_THREAD; ++it) {
    const int q = qbase + it*THREADS;
    float4 v;
#if ATHENA_ASYNC_LDS
    if (it + 1 < F4_PER_THREAD) __builtin_amdgcn_s_wait_asynccnt(1);
    else                        __builtin_amdgcn_s_wait_asynccnt(0);
    __syncthreads();
    v = ((const float4*)tile[it & 1])[threadIdx.x];
#else
    v = ((const float4*)rowbase)[q];
    if (it + 2 < F4_PER_THREAD)
      __builtin_prefetch(rowbase + (size_t)(q + 2*THREADS)*4, 0, 1);
#endif
    float lv[4]; uint32_t lb[4];
    athena_loss4(v, q, d0, h0, w0, tpatch, lv, lb);
#pragma unroll
    for (int j = 0; j < 4; ++j)
      if (lb[j] > thr) acc += (double)lv[j];
#if ATHENA_ASYNC_LDS
    __syncthreads();
    if (it + 2 < F4_PER_THREAD)
      athena_async16(rowbase + (size_t)(q + 2*THREADS)*4, &tile[it & 1][threadIdx.x*4]);
#endif
  }
  sred[threadIdx.x] = acc;
  __syncthreads();
  for (int s = THREADS/2; s > 0; s >>= 1) {
    if (threadIdx.x < s) sred[threadIdx.x] += sred[threadIdx.x + s];
    __syncthreads();
  }
  if (threadIdx.x == 0) atomicAdd(&row_sum[r], sred[0]);
}

// ---- Final sum reading cached bit patterns
__global__ void __launch_bounds__(THREADS)
athena_sum_bits_kernel(const unsigned int* __restrict__ bits_arr,
                       const unsigned int* __restrict__ state,
                       double* __restrict__ row_sum)
{
  __shared__ double sred[THREADS];
  const int r = blockIdx.y;
  const unsigned int thr = state[r*4+2];
  const uint4* base = (const uint4*)(bits_arr + (size_t)r*ROWLEN);
  const int qbase = blockIdx.x * F4_PER_BLOCK + threadIdx.x;
  double acc = 0.0;
  for (int it = 0; it < F4_PER_THREAD; ++it) {
    uint4 u = base[qbase + it*THREADS];
    unsigned int b4[4] = {u.x, u.y, u.z, u.w};
#pragma unroll
    for (int j = 0; j < 4; ++j)
      if (b4[j] > thr) acc += (double)__uint_as_float(b4[j]);
  }
  sred[threadIdx.x] = acc;
  __syncthreads();
  for (int s = THREADS/2; s > 0; s >>= 1) {
    if (threadIdx.x < s) sred[threadIdx.x] += sred[threadIdx.x + s];
    __syncthreads();
  }
  if (threadIdx.x == 0) atomicAdd(&row_sum[r], sred[0]);
}

// ---- Finalize: per-row total = sum_gt + k4 * thr_value; reduce 16 rows.
// WMMA trick: A = all-ones 16x4, B holds the 16 row totals in any bijective
// slot mapping -> sum over D row 0 equals the grand total regardless of the
// exact B register striping.
typedef __attribute__((ext_vector_type(2))) float athena_v2f;
typedef __attribute__((ext_vector_type(8))) float athena_v8f;

__global__ void athena_finalize_kernel(const unsigned int* __restrict__ state,
                                       const double* __restrict__ row_sum,
                                       float* __restrict__ out)
{
  const int lane = threadIdx.x;   // launched with exactly 32 threads (1 wave)
  double s = 0.0;
  if (lane < NROWS) {
    s = row_sum[lane];
    unsigned int thr = state[lane*4+2];
    unsigned int k4  = state[lane*4+1];
    s += (double)k4 * (double)__uint_as_float(thr);   // tie contribution
  }
#if ATHENA_WMMA
  athena_v2f a;  a[0] = 1.f; a[1] = 1.f;              // all-ones A (16x4)
  athena_v2f bm; bm[0] = (lane < NROWS) ? (float)s : 0.f; bm[1] = 0.f;
  athena_v8f c = {0.f,0.f,0.f,0.f,0.f,0.f,0.f,0.f};
  c = __builtin_amdgcn_wmma_f32_16x16x4_f32(false, a, false, bm, (short)0, c,
                                            false, false);
  float x = (lane < 16) ? c[0] : 0.f;                 // D[0][n] lives in lane n, vgpr0
  for (int off = 16; off > 0; off >>= 1) x += __shfl_down(x, off, 32);
  if (lane == 0) *out = x / (16.0f * (float)NTOP);
#else
  __shared__ double sh[32];
  sh[lane] = s;
  __syncthreads();
  if (lane == 0) {
    double tot = 0.0;
    for (int i = 0; i < NROWS; ++i) tot += sh[i];
    *out = (float)(tot / (16.0 * (double)NTOP));
  }
#endif
}

extern "C" void kernel_launch(void* const* d_in, const int* in_sizes, int n_in,
                              void* d_out, int out_size, void* d_ws, size_t ws_size,
                              hipStream_t stream)
{
  (void)in_sizes; (void)n_in; (void)out_size;
  const float* net     = (const float*)d_in[0];
  const float* tstruct = (const float*)d_in[1];
  const int*   bboxes  = (const int*)d_in[2];
  float* out = (float*)d_out;

  unsigned char* ws = (unsigned char*)d_ws;
  const size_t HIST_BYTES  = (size_t)NROWS * NBIN * 4;           // 262144
  const size_t STATE_OFF   = HIST_BYTES;                         // 16*4 u32 (+pad)
  const size_t ROWSUM_OFF  = HIST_BYTES + 256;                   // 16 doubles
  const size_t BITS_OFF    = HIST_BYTES + 256 + 128;             // 151 MB cache
  const size_t BITS_BYTES  = (size_t)NROWS * ROWLEN * 4;

  unsigned int* hist    = (unsigned int*)ws;
  unsigned int* state   = (unsigned int*)(ws + STATE_OFF);
  double*       row_sum = (double*)(ws + ROWSUM_OFF);
  unsigned int* bits    = (unsigned int*)(ws + BITS_OFF);

  const bool cached = ws_size >= BITS_OFF + BITS_BYTES;   // 151MB fits L2 (192MB)

  (void)hipMemsetAsync(ws, 0, BITS_OFF, stream);   // zero hist/state/row_sum only

  dim3 grid(BLOCKS_PER_ROW, NROWS);
  if (cached) {
    athena_hist_kernel<0, true><<<grid, THREADS, 0, stream>>>(net, tstruct, bboxes, hist, state, bits);
    athena_scan_kernel<<<NROWS, THREADS, 0, stream>>>(hist, state, 0);
    athena_hist_bits_kernel<1><<<grid, THREADS, 0, stream>>>(bits, hist, state);
    athena_scan_kernel<<<NROWS, THREADS, 0, stream>>>(hist, state, 1);
    athena_hist_bits_kernel<2><<<grid, THREADS, 0, stream>>>(bits, hist, state);
    athena_scan_kernel<<<NROWS, THREADS, 0, stream>>>(hist, state, 2);
    athena_sum_bits_kernel<<<grid, THREADS, 0, stream>>>(bits, state, row_sum);
  } else {
    athena_hist_kernel<0, false><<<grid, THREADS, 0, stream>>>(net, tstruct, bboxes, hist, state, nullptr);
    athena_scan_kernel<<<NROWS, THREADS, 0, stream>>>(hist, state, 0);
    athena_hist_kernel<1, false><<<grid, THREADS, 0, stream>>>(net, tstruct, bboxes, hist, state, nullptr);
    athena_scan_kernel<<<NROWS, THREADS, 0, stream>>>(hist, state, 1);
    athena_hist_kernel<2, false><<<grid, THREADS, 0, stream>>>(net, tstruct, bboxes, hist, state, nullptr);
    athena_scan_kernel<<<NROWS, THREADS, 0, stream>>>(hist, state, 2);
    athena_sum_kernel<<<grid, THREADS, 0, stream>>>(net, tstruct, bboxes, state, row_sum);
  }
  athena_finalize_kernel<<<1, 32, 0, stream>>>(state, row_sum, out);
}